// FastRGCNConv_33500744909322
// MI455X (gfx1250) — compile-verified
//
#include <hip/hip_runtime.h>
#include <hip/hip_bf16.h>

typedef __attribute__((ext_vector_type(2))) float v2f;
typedef __attribute__((ext_vector_type(8))) float v8f;

#define N_NODES 100000
#define N_EDGES 600000
#define D 128
#define N_REL 16
#define KTOT 384            // 128 (x) + 128 (y0) + 128 (y1)
#define KC   64             // K-chunk staged in LDS
#define KPAIRS (KC / 2)     // 32 K-pair rows per chunk
#define P_STRIDE 144        // float2 per pair-row: 2*144 mod 64 == 32 -> conflict-free b64 reads
#define STRIPS (N_NODES / 16)   // 6250 exact
#define WAVES_PER_BLOCK 8

// ---------------- zero workspace ----------------
__global__ void rgcn_zero_kernel(float4* __restrict__ p, long n4) {
    long i = (long)blockIdx.x * blockDim.x + threadIdx.x;
    long stride = (long)gridDim.x * blockDim.x;
    for (; i < n4; i += stride) p[i] = make_float4(0.f, 0.f, 0.f, 0.f);
}

// ---------------- weighted edge scatter (basis-folded) ----------------
// y_b[tail] += base_weights[type, b] * x[head]   for b in {0,1}
__global__ void rgcn_scatter_kernel(const float* __restrict__ x,
                                    const int* __restrict__ heads,
                                    const int* __restrict__ tails,
                                    const int* __restrict__ etype,
                                    const float* __restrict__ bw,   // [N_REL, 2]
                                    float* __restrict__ y0,
                                    float* __restrict__ y1) {
    const int e = blockIdx.x * 2 + (threadIdx.x >> 7);   // 2 edges per 256-thread block
    if (e >= N_EDGES) return;
    const int d = threadIdx.x & (D - 1);
    const int h = heads[e];
    const int t = tails[e];
    const int r = etype[e];
    const float c0 = bw[r * 2 + 0];
    const float c1 = bw[r * 2 + 1];
    const float v = x[(long)h * D + d];
    // native fp32 atomic -> global_atomic_add_f32 (L2-resident RMW)
    unsafeAtomicAdd(&y0[(long)t * D + d], c0 * v);
    unsafeAtomicAdd(&y1[(long)t * D + d], c1 * v);
}

// ---- fused GEMM: out = [x|y0|y1] @ [Wself;B0;B1] via V_WMMA_F32_16X16X4_F32 ----
// LDS weight layout: K-pair interleaved, ldsw2[(k>>1)*P_STRIDE + n] = {W[k][n], W[k+1][n]}
// so one B fragment == one aligned ds_load_b64 straight into a WMMA operand pair.
__global__ void __launch_bounds__(256, 2)
rgcn_gemm_kernel(const float* __restrict__ x,
                 const float* __restrict__ y0,
                 const float* __restrict__ y1,
                 const float* __restrict__ wself,   // [128,128]
                 const float* __restrict__ bases,   // [2,128,128]
                 float* __restrict__ out) {
    __shared__ v2f ldsw2[KPAIRS * P_STRIDE];   // 36 KB

    const int lane  = threadIdx.x & 31;
    const int wave  = threadIdx.x >> 5;
    const int strip = blockIdx.x * WAVES_PER_BLOCK + wave;   // 16-row M strip
    const bool active = (strip < STRIPS);                    // wave-uniform

    const int mlo   = lane & 15;        // M within strip for A / N for B,C
    const int khp   = lane >> 4;        // half-wave selector (K pair offset)
    const int khalf = khp * 2;

    v8f acc[8];
#pragma unroll
    for (int t = 0; t < 8; ++t) acc[t] = v8f{};

    // K chunks: c=0,1 -> x/Wself ; c=2,3 -> y0/bases[0] ; c=4,5 -> y1/bases[1]
    for (int c = 0; c < KTOT / KC; ++c) {
        const float* wsrc = (c < 2) ? wself : (c < 4 ? bases : bases + D * D);
        const float* asrc = (c < 2) ? x     : (c < 4 ? y0    : y1);
        const int krow0 = (c & 1) * KC;   // row offset inside the 128-row source

        // cooperative LDS fill: each unit = (pair-row, 4-col group):
        // load float4 from rows k and k+1, write 4 interleaved ds_store_b64
#pragma unroll
        for (int i = 0; i < (KPAIRS * (D / 4)) / 256; ++i) {   // 1024 units / 256 thr = 4
            const int u  = i * 256 + threadIdx.x;
            const int kp = u >> 5;               // pair-row 0..31
            const int g  = (u & 31) * 4;         // col group base
            const float4 wa = *(const float4*)(wsrc + (long)(krow0 + 2 * kp + 0) * D + g);
            const float4 wb = *(const float4*)(wsrc + (long)(krow0 + 2 * kp + 1) * D + g);
            v2f* dst = &ldsw2[kp * P_STRIDE + g];
            dst[0] = v2f{wa.x, wb.x};
            dst[1] = v2f{wa.y, wb.y};
            dst[2] = v2f{wa.z, wb.z};
            dst[3] = v2f{wa.w, wb.w};
        }
        __syncthreads();

        if (active) {
            const float* arow = asrc + (long)(strip * 16 + mlo) * D + krow0;
            const v2f* bbase = &ldsw2[khp * P_STRIDE + mlo];
#pragma unroll 4
            for (int kk = 0; kk < KC; kk += 4) {
                // A fragment 16x4: lane (M=mlo, K=kk+khalf+{0,1}) -- one global b64
                const v2f a = *(const v2f*)(arow + kk + khalf);
                const v2f* brow = bbase + (kk >> 1) * P_STRIDE;
#pragma unroll
                for (int t = 0; t < 8; ++t) {
                    // B fragment 4x16: lane (K=kk+khalf+{0,1}, N=t*16+mlo) -- one ds_load_b64
                    const v2f b = brow[t * 16];
                    acc[t] = __builtin_amdgcn_wmma_f32_16x16x4_f32(
                        false, a, false, b, (short)0, acc[t], false, false);
                }
            }
        }
        __syncthreads();
    }

    if (active) {
        // C/D layout: lane L, VGPR v -> (M = v + 8*(L>>4), N = L%16)
        const int mbase = strip * 16 + 8 * khp;
#pragma unroll
        for (int t = 0; t < 8; ++t) {
#pragma unroll
            for (int v = 0; v < 8; ++v)
                out[(long)(mbase + v) * D + t * 16 + mlo] = acc[t][v];
        }
    }
}

extern "C" void kernel_launch(void* const* d_in, const int* in_sizes, int n_in,
                              void* d_out, int out_size, void* d_ws, size_t ws_size,
                              hipStream_t stream) {
    const float* x      = (const float*)d_in[0];                 // [N,128]
    const int*   eidx   = (const int*)d_in[1];                   // [2,E]
    const int*   etype  = (const int*)d_in[2];                   // [E]
    const float* wself  = (const float*)d_in[3];                 // [128,128]
    const float* bases  = (const float*)d_in[4];                 // [2,128,128]
    const float* bw     = (const float*)d_in[5];                 // [16,2]
    float* out = (float*)d_out;

    float* y0 = (float*)d_ws;                                    // [N,128]
    float* y1 = y0 + (long)N_NODES * D;                          // [N,128]

    // 1) zero basis accumulators (atomics accumulate -> must re-zero every call)
    const long n4 = (long)2 * N_NODES * D / 4;
    rgcn_zero_kernel<<<8192, 256, 0, stream>>>((float4*)d_ws, n4);

    // 2) basis-folded weighted scatter over edges
    rgcn_scatter_kernel<<<N_EDGES / 2, 256, 0, stream>>>(
        x, eidx, eidx + N_EDGES, etype, bw, y0, y1);

    // 3) fused [N,384] x [384,128] GEMM on fp32 WMMA
    const int grid = (STRIPS + WAVES_PER_BLOCK - 1) / WAVES_PER_BLOCK;   // 782
    rgcn_gemm_kernel<<<grid, 256, 0, stream>>>(x, y0, y1, wself, bases, out);
}